// AverageItemProfile_34591666602703
// MI455X (gfx1250) — compile-verified
//
#include <hip/hip_runtime.h>

typedef float v4f __attribute__((ext_vector_type(4)));

#define MIN_R 1.0f
#define MAX_R 5.0f

// ---------------------------------------------------------------------------
// Phase 1 (placed first in the module so the disasm snippet shows the atomic
// codegen): scatter-add ratings and counts with hardware f32 atomics.
// Batch streams read non-temporally so the 36MB item table stays L2-resident.
// ---------------------------------------------------------------------------
__global__ __launch_bounds__(256) void aip_phase1_scatter(
    const int* __restrict__ ids, const float* __restrict__ ratings,
    float* __restrict__ ip_ws, float* __restrict__ cnt_ws, int B) {
  int b = blockIdx.x * blockDim.x + threadIdx.x;
  if (b >= B) return;
  int id = __builtin_nontemporal_load(ids + b);
  const v4f* r = (const v4f*)(ratings + (size_t)b * 8);
  v4f lo = __builtin_nontemporal_load(r);
  v4f hi = __builtin_nontemporal_load(r + 1);
  float* dst = ip_ws + (size_t)id * 8;
  atomicAdd(dst + 0, lo.x);
  atomicAdd(dst + 1, lo.y);
  atomicAdd(dst + 2, lo.z);
  atomicAdd(dst + 3, lo.w);
  atomicAdd(dst + 4, hi.x);
  atomicAdd(dst + 5, hi.y);
  atomicAdd(dst + 6, hi.z);
  atomicAdd(dst + 7, hi.w);
  atomicAdd(cnt_ws + id, 1.0f);
}

// ---------------------------------------------------------------------------
// Phase 2: gather (L2-resident table), fold in the division, clamp, dot with
// weights, non-temporal store of predictions.
// ---------------------------------------------------------------------------
__global__ __launch_bounds__(256) void aip_phase2_predict(
    const int* __restrict__ ids, const float* __restrict__ weights,
    const float* __restrict__ ip_ws, const float* __restrict__ cnt_ws,
    float* __restrict__ out, int B) {
  int b = blockIdx.x * blockDim.x + threadIdx.x;
  if (b >= B) return;
  int id = __builtin_nontemporal_load(ids + b);
  const v4f* w4 = (const v4f*)(weights + (size_t)b * 8);
  v4f w0 = __builtin_nontemporal_load(w4);
  v4f w1 = __builtin_nontemporal_load(w4 + 1);
  const v4f* p = (const v4f*)(ip_ws + (size_t)id * 8);
  v4f p0 = p[0];
  v4f p1 = p[1];
  float cnt = cnt_ws[id];
  float inv = 1.0f / fmaxf(cnt, 1.0f);
  float acc = 0.0f;
#pragma unroll
  for (int a = 0; a < 4; ++a) {
    float r0 = fminf(fmaxf(p0[a] * inv, MIN_R), MAX_R);
    float r1 = fminf(fmaxf(p1[a] * inv, MIN_R), MAX_R);
    acc = fmaf(w0[a], r0, acc);
    acc = fmaf(w1[a], r1, acc);
  }
  __builtin_nontemporal_store(acc, out + b);
}

// ---------------------------------------------------------------------------
// Phase 0: un-average into workspace.
//   ip_ws[v][0..7] = params[v][0..7] * counters[v];  cnt_ws[v] = counters[v]
// One thread per item row; 2x B128 loads + 2x B128 stores (verified codegen:
// global_load_b128 / v_pk_mul_f32 / global_store_b128).
// ---------------------------------------------------------------------------
__global__ __launch_bounds__(256) void aip_phase0_unavg(
    const float* __restrict__ params, const float* __restrict__ counters,
    float* __restrict__ ip_ws, float* __restrict__ cnt_ws, int V) {
  int v = blockIdx.x * blockDim.x + threadIdx.x;
  if (v >= V) return;
  float c = counters[v];
  const v4f* p = (const v4f*)(params + (size_t)v * 8);
  v4f lo = p[0];
  v4f hi = p[1];
  lo *= c;
  hi *= c;
  v4f* o = (v4f*)(ip_ws + (size_t)v * 8);
  o[0] = lo;
  o[1] = hi;
  cnt_ws[v] = c;
}

extern "C" void kernel_launch(void* const* d_in, const int* in_sizes, int n_in,
                              void* d_out, int out_size, void* d_ws, size_t ws_size,
                              hipStream_t stream) {
  const float* params   = (const float*)d_in[0];  // [V,8]
  const float* counters = (const float*)d_in[1];  // [V]
  const int*   ids      = (const int*)d_in[2];    // [B]
  const float* weights  = (const float*)d_in[3];  // [B,8]
  const float* ratings  = (const float*)d_in[4];  // [B,8]
  float* out = (float*)d_out;                     // [B]

  const int V = in_sizes[1];
  const int B = in_sizes[2];

  // Workspace layout: [V*8] f32 un-averaged profiles, then [V] f32 counts.
  float* ip_ws  = (float*)d_ws;
  float* cnt_ws = ip_ws + (size_t)V * 8;
  (void)ws_size;  // needs V*36 bytes (~36 MB)

  const int TB = 256;
  dim3 block(TB);
  dim3 gridV((V + TB - 1) / TB);
  dim3 gridB((B + TB - 1) / TB);

  aip_phase0_unavg<<<gridV, block, 0, stream>>>(params, counters, ip_ws, cnt_ws, V);
  aip_phase1_scatter<<<gridB, block, 0, stream>>>(ids, ratings, ip_ws, cnt_ws, B);
  aip_phase2_predict<<<gridB, block, 0, stream>>>(ids, weights, ip_ws, cnt_ws, out, B);
}